// SpectralDecompositionDeformableAttention_1623497638375
// MI455X (gfx1250) — compile-verified
//
#include <hip/hip_runtime.h>
#include <hip/hip_bf16.h>

// ---------------------------------------------------------------------------
// SpectralDecompositionDeformableAttention for MI455X (gfx1250, wave32).
//
// Roofline: ~20 GFLOP dense + ~0.25 GB HBM traffic -> memory/latency bound at
// 23.3 TB/s, so we keep full fp32 precision and use V_WMMA_F32_16X16X4_F32
// for every GEMM (accuracy free, matrix pipe exercised, weights live in L2).
// ---------------------------------------------------------------------------

typedef __attribute__((ext_vector_type(2))) float v2f;
typedef __attribute__((ext_vector_type(8))) float v8f;

#define NB 8
#define NQ 900
#define NC 256
#define NH 8
#define NL 4
#define NP 4
#define ND 32
#define NS 13294
#define BQ (NB * NQ)          // 7200
#define MV (NB * NS)          // 106352

// ---------------------------------------------------------------------------
// Generic WMMA fp32 GEMM:  C[M,N] = A[M,K] * W[K,N] + bias[N]  (opt. ReLU)
// Requirements: M % 16 == 0, N % 64 == 0, K % 4 == 0 (all satisfied here).
// Each wave computes a 16x64 strip: 4 accumulators, A fragment reused 4x.
//
// Wave32 fragment layouts (ISA 7.12.2):
//   A (16x4 f32, 2 VGPR):  lanes 0-15 -> rows, {v0,v1}={K,K+1}; lanes 16-31 -> {K+2,K+3}
//   B (4x16 f32, 2 VGPR):  v0: lanes0-15=row K cols, lanes16-31=row K+1; v1: rows K+2,K+3
//   C/D (16x16 f32, 8 VGPR): v_i: lanes0-15 -> M=i, lanes16-31 -> M=i+8
// ---------------------------------------------------------------------------
__global__ __launch_bounds__(256) void gemm_wmma_f32(
    const float* __restrict__ A, const float* __restrict__ W,
    const float* __restrict__ bias, float* __restrict__ C,
    int M, int N, int K, int relu)
{
    const int lane  = threadIdx.x & 31;
    const int wave  = threadIdx.x >> 5;
    const int wid   = blockIdx.x * (blockDim.x >> 5) + wave;
    const int tilesN = N >> 6;                       // 64-wide strips
    const int total  = (M >> 4) * tilesN;
    if (wid >= total) return;                        // wave-uniform exit

    const int tileM = (wid / tilesN) << 4;
    const int tileN = (wid % tilesN) << 6;

    v8f acc0 = {}, acc1 = {}, acc2 = {}, acc3 = {};

    const int rowA  = tileM + (lane & 15);
    const int kHalf = (lane >> 4) << 1;              // 0 or 2 (A lanes 16-31 hold K+2,K+3)
    const int colB  = lane & 15;
    const int kOff  = lane >> 4;                     // 0 or 1 (B lanes 16-31 hold next K row)

    const float* __restrict__ Arow = A + (size_t)rowA * K;

    for (int k = 0; k < K; k += 4) {
        v2f a;
        a.x = Arow[k + kHalf];
        a.y = Arow[k + kHalf + 1];

        const float* __restrict__ W0 = W + (size_t)(k + kOff) * N + tileN + colB;
        const float* __restrict__ W2 = W0 + 2 * (size_t)N;
        v2f b0, b1, b2, b3;
        b0.x = W0[0];   b0.y = W2[0];
        b1.x = W0[16];  b1.y = W2[16];
        b2.x = W0[32];  b2.y = W2[32];
        b3.x = W0[48];  b3.y = W2[48];

        acc0 = __builtin_amdgcn_wmma_f32_16x16x4_f32(false, a, false, b0, (short)0, acc0, false, false);
        acc1 = __builtin_amdgcn_wmma_f32_16x16x4_f32(false, a, false, b1, (short)0, acc1, false, false);
        acc2 = __builtin_amdgcn_wmma_f32_16x16x4_f32(false, a, false, b2, (short)0, acc2, false, false);
        acc3 = __builtin_amdgcn_wmma_f32_16x16x4_f32(false, a, false, b3, (short)0, acc3, false, false);
    }

    const int rowBase = tileM + ((lane >> 4) << 3);  // +8 for lanes 16-31
    const int colLane = lane & 15;
    #pragma unroll
    for (int t = 0; t < 4; ++t) {
        v8f acc = (t == 0) ? acc0 : (t == 1) ? acc1 : (t == 2) ? acc2 : acc3;
        const int col = tileN + t * 16 + colLane;
        const float bv = bias[col];
        #pragma unroll
        for (int i = 0; i < 8; ++i) {
            float x = acc[i] + bv;
            if (relu) x = fmaxf(x, 0.0f);
            C[(size_t)(rowBase + i) * N + col] = x;
        }
    }
}

// ---------------------------------------------------------------------------
// Build padded [q | object_shape | 0 0] matrix: (BQ x 260), K padded 258->260
// ---------------------------------------------------------------------------
__global__ void build_qshape(const float* __restrict__ q,
                             const float* __restrict__ obj,
                             float* __restrict__ qs)
{
    int i = blockIdx.x * blockDim.x + threadIdx.x;
    if (i >= BQ * 260) return;
    int r = i / 260, c = i % 260;
    float v;
    if (c < NC)       v = q[r * NC + c];
    else if (c < 258) v = obj[r * 2 + (c - 256)];
    else              v = 0.0f;
    qs[i] = v;
}

// Zero-padded w_off1: (258,256) -> (260,256)
__global__ void build_woff1(const float* __restrict__ w, float* __restrict__ wp)
{
    int i = blockIdx.x * blockDim.x + threadIdx.x;
    if (i >= 260 * NC) return;
    int r = i / NC, c = i % NC;
    wp[i] = (r < 258) ? w[r * NC + c] : 0.0f;
}

// ---------------------------------------------------------------------------
// Fused spectral softmax + bilinear sampling. One wave32 per (b,q,h), lane=d.
// ---------------------------------------------------------------------------
__global__ __launch_bounds__(256) void sample_agg_kernel(
    const float* __restrict__ qbuf, const float* __restrict__ vbuf,
    const float* __restrict__ offs, const float* __restrict__ axb,
    const float* __restrict__ ayb,  const float* __restrict__ refp,
    const float* __restrict__ obj,  const float* __restrict__ basis_x,
    const float* __restrict__ basis_y, float* __restrict__ agg)
{
    const int lane = threadIdx.x & 31;
    const int wid  = (blockIdx.x * blockDim.x + threadIdx.x) >> 5;
    if (wid >= NB * NQ * NH) return;

    const int h  = wid % NH;
    const int bq = wid / NH;          // b*Q + q
    const int b  = bq / NQ;
    const int d  = lane;

    // spec = (q_h . basis_x) * (q_h . basis_y)   -- full wave32 reduction
    float qv = qbuf[(size_t)bq * NC + h * ND + d];
    float sx = qv * basis_x[h * ND + d];
    float sy = qv * basis_y[h * ND + d];
    #pragma unroll
    for (int m = 16; m >= 1; m >>= 1) {
        sx += __shfl_xor(sx, m, 32);
        sy += __shfl_xor(sy, m, 32);
    }
    const float spec = sx * sy;

    // softmax over L*P = 16 logits (redundant per lane; scalar-cache reads)
    float logit[16];
    float mx = -1e30f;
    #pragma unroll
    for (int i = 0; i < 16; ++i) {
        float l = axb[(size_t)bq * (NH * NL * NP) + h * 16 + i] *
                  ayb[(size_t)bq * (NH * NL * NP) + h * 16 + i] + spec;
        logit[i] = l;
        mx = fmaxf(mx, l);
    }
    float denom = 0.0f;
    #pragma unroll
    for (int i = 0; i < 16; ++i) { logit[i] = __expf(logit[i] - mx); denom += logit[i]; }
    const float inv = 1.0f / denom;

    const float osx = obj[bq * 2 + 0];
    const float osy = obj[bq * 2 + 1];

    const int   Wl_[4] = {100, 50, 25, 13};
    const int   Hl_[4] = {100, 50, 25, 13};
    const int   st_[4] = {0, 10000, 12500, 13125};

    float acc = 0.0f;
    #pragma unroll
    for (int i = 0; i < 16; ++i) {
        const int l  = i >> 2;
        const int Wl = Wl_[l], Hl = Hl_[l], st = st_[l];

        const float rx = refp[((size_t)bq * NL + l) * 2 + 0];
        const float ry = refp[((size_t)bq * NL + l) * 2 + 1];
        const float ox = offs[(size_t)bq * NC + h * 32 + i * 2 + 0] * osx;
        const float oy = offs[(size_t)bq * NC + h * 32 + i * 2 + 1] * osy;
        const float lx = fminf(fmaxf(rx + ox, 0.0f), 1.0f);
        const float ly = fminf(fmaxf(ry + oy, 0.0f), 1.0f);

        const float x  = lx * (float)Wl - 0.5f;
        const float y  = ly * (float)Hl - 0.5f;
        const float x0 = floorf(x), y0 = floorf(y);
        const float fx = x - x0,    fy = y - y0;
        const int x0i = (int)x0,    y0i = (int)y0;
        const float wgt = logit[i] * inv;

        #pragma unroll
        for (int c = 0; c < 4; ++c) {
            const int dx = c & 1, dy = c >> 1;
            const int xi = x0i + dx, yi = y0i + dy;
            const float cw = (dx ? fx : 1.0f - fx) * (dy ? fy : 1.0f - fy);
            const float valid = ((xi >= 0) & (xi < Wl) & (yi >= 0) & (yi < Hl)) ? 1.0f : 0.0f;
            const int xc = min(max(xi, 0), Wl - 1);
            const int yc = min(max(yi, 0), Hl - 1);
            const int sidx = st + yc * Wl + xc;
            // coalesced: 32 lanes read 32 consecutive floats (one head slice)
            const float g = vbuf[((size_t)b * NS + sidx) * NC + h * ND + d];
            acc += wgt * cw * valid * g;
        }
    }
    agg[(size_t)bq * NC + h * ND + d] = acc;
}

// ---------------------------------------------------------------------------
extern "C" void kernel_launch(void* const* d_in, const int* in_sizes, int n_in,
                              void* d_out, int out_size, void* d_ws, size_t ws_size,
                              hipStream_t stream)
{
    const float* query  = (const float*)d_in[0];
    const float* refp   = (const float*)d_in[1];
    const float* inpf   = (const float*)d_in[2];
    const float* obj    = (const float*)d_in[3];
    // d_in[4], d_in[5]: spatial shapes / level starts (compile-time constants here)
    const float* w_q    = (const float*)d_in[6];   const float* b_q    = (const float*)d_in[7];
    const float* w_v    = (const float*)d_in[8];   const float* b_v    = (const float*)d_in[9];
    const float* w_o    = (const float*)d_in[10];  const float* b_o    = (const float*)d_in[11];
    const float* w_off1 = (const float*)d_in[12];  const float* b_off1 = (const float*)d_in[13];
    const float* w_off2 = (const float*)d_in[14];  const float* b_off2 = (const float*)d_in[15];
    const float* w_ax   = (const float*)d_in[16];  const float* b_ax   = (const float*)d_in[17];
    const float* w_ay   = (const float*)d_in[18];  const float* b_ay   = (const float*)d_in[19];
    const float* bx     = (const float*)d_in[20];  const float* by     = (const float*)d_in[21];

    float* ws   = (float*)d_ws;
    float* qbuf = ws;  ws += (size_t)BQ * NC;       // 7200 x 256
    float* vbuf = ws;  ws += (size_t)MV * NC;       // 106352 x 256
    float* qsp  = ws;  ws += (size_t)BQ * 260;      // padded [q|shape|0]
    float* w1p  = ws;  ws += (size_t)260 * NC;      // padded w_off1
    float* hid  = ws;  ws += (size_t)BQ * NC;
    float* offs = ws;  ws += (size_t)BQ * NC;
    float* axb  = ws;  ws += (size_t)BQ * 128;
    float* ayb  = ws;  ws += (size_t)BQ * 128;
    float* agg  = ws;                               // 7200 x 256

    auto gemm = [&](const float* A, const float* W, const float* bias,
                    float* C, int M, int N, int K, int relu) {
        int tiles  = (M >> 4) * (N >> 6);
        int blocks = (tiles + 7) / 8;               // 8 waves per 256-thread block
        hipLaunchKernelGGL(gemm_wmma_f32, dim3(blocks), dim3(256), 0, stream,
                           A, W, bias, C, M, N, K, relu);
    };

    // 1) q = query @ w_q + b_q
    gemm(query, w_q, b_q, qbuf, BQ, NC, NC, 0);
    // 2) v = input_flatten @ w_v + b_v   (the big one: M = 106352)
    gemm(inpf, w_v, b_v, vbuf, MV, NC, NC, 0);
    // 3) padded concat + padded w_off1
    {
        int n1 = BQ * 260;
        hipLaunchKernelGGL(build_qshape, dim3((n1 + 255) / 256), dim3(256), 0, stream, qbuf, obj, qsp);
        int n2 = 260 * NC;
        hipLaunchKernelGGL(build_woff1, dim3((n2 + 255) / 256), dim3(256), 0, stream, w_off1, w1p);
    }
    // 4) hid = relu(q_shape @ w_off1 + b_off1)   (K = 260 padded)
    gemm(qsp, w1p, b_off1, hid, BQ, NC, 260, 1);
    // 5) offsets_raw = hid @ w_off2 + b_off2
    gemm(hid, w_off2, b_off2, offs, BQ, NC, NC, 0);
    // 6) attn_x / attn_y   (N = 128)
    gemm(qbuf, w_ax, b_ax, axb, BQ, 128, NC, 0);
    gemm(qbuf, w_ay, b_ay, ayb, BQ, 128, NC, 0);
    // 7) fused softmax + bilinear sample + aggregate
    {
        int waves  = NB * NQ * NH;                  // 57600
        int blocks = waves / 8;                     // 8 waves per block
        hipLaunchKernelGGL(sample_agg_kernel, dim3(blocks), dim3(256), 0, stream,
                           qbuf, vbuf, offs, axb, ayb, refp, obj, bx, by, agg);
    }
    // 8) out = agg @ w_o + b_o
    gemm(agg, w_o, b_o, (float*)d_out, BQ, NC, NC, 0);
}